// EGNN_Edge_9698036154373
// MI455X (gfx1250) — compile-verified
//
#include <hip/hip_runtime.h>

#define SDIM 128
#define EDIM 128
#define HID  64
#define MSG_IN 386
#define SCAL_IN 192
#define DEPTH 4
#define CUTOFF 5.0f
#define EPS_D 1e-6f

typedef __attribute__((ext_vector_type(2))) float v2f;
typedef __attribute__((ext_vector_type(8))) float v8f;

#define WAVES 4
#define BSTRIDE 132   // 16x128 staging rows, padded; row = 528B (16B multiple)
#define MSTRIDE 68    // 16x64 transpose rows, padded; row = 272B (16B multiple)

static __device__ __forceinline__ v8f wmma4(v2f a, v2f b, v8f c) {
    // V_WMMA_F32_16X16X4_F32 : D = A(16x4) * B(4x16) + C, all fp32
    return __builtin_amdgcn_wmma_f32_16x16x4_f32(false, a, false, b, (short)0, c, false, false);
}
static __device__ __forceinline__ float silu_f(float x) {
    return x / (1.0f + __expf(-x));
}

__global__ void zero_kernel(float* __restrict__ p, long long n) {
    long long i = (long long)blockIdx.x * blockDim.x + threadIdx.x;
    if (i < n) p[i] = 0.0f;
}

__global__ void deg_kernel(const int* __restrict__ dst, float* __restrict__ deg, int E) {
    int i = blockIdx.x * blockDim.x + threadIdx.x;
    if (i < E) atomicAdd(&deg[dst[i]], 1.0f);
}

// ---------------------------------------------------------------------------
// Edge kernel: one 16-edge tile per wave.
//   m = silu(silu(h@We1+be1)@We2+be2) * cw ;  e += m@Weo+beo
//   wv = m@Wv+bv ;  vacc[dst] += rel*wv ;  agg[dst] += m
// ---------------------------------------------------------------------------
__global__ __launch_bounds__(WAVES * 32)
void edge_kernel(const float* __restrict__ s, const float* __restrict__ v,
                 float* __restrict__ e,
                 const int* __restrict__ src, const int* __restrict__ dst,
                 const float* __restrict__ ea_d,
                 const float* __restrict__ We1, const float* __restrict__ be1,
                 const float* __restrict__ We2, const float* __restrict__ be2,
                 const float* __restrict__ Weo, const float* __restrict__ beo,
                 const float* __restrict__ Wv,  const float* __restrict__ bv,
                 float* __restrict__ vacc, float* __restrict__ agg, int ntiles)
{
    __shared__ __align__(16) float buf[WAVES][16][BSTRIDE]; // staged A rows (s_src / s_dst / e)
    __shared__ __align__(16) float mM [WAVES][16][MSTRIDE]; // hidden activations, row-major
    __shared__ float sdist[WAVES][16], sead[WAVES][16], scw[WAVES][16];
    __shared__ float srel[WAVES][16][3];
    __shared__ int   ssrc[WAVES][16], sdstn[WAVES][16];

    const int w    = threadIdx.x >> 5;
    const int lane = threadIdx.x & 31;
    const int tile = blockIdx.x * WAVES + w;
    if (tile >= ntiles) return;          // wave-uniform
    const int half = lane >> 4;
    const int n    = lane & 15;          // also A-row index m for this lane

    if (lane < 16) {
        int eid = tile * 16 + lane;
        int a = src[eid], b = dst[eid];
        ssrc[w][lane]  = a;
        sdstn[w][lane] = b;
        float rx = v[a*3+0] - v[b*3+0];
        float ry = v[a*3+1] - v[b*3+1];
        float rz = v[a*3+2] - v[b*3+2];
        srel[w][lane][0] = rx; srel[w][lane][1] = ry; srel[w][lane][2] = rz;
        float d = sqrtf(rx*rx + ry*ry + rz*rz + EPS_D);
        sdist[w][lane] = d;
        sead[w][lane]  = ea_d[eid];
        scw[w][lane]   = (d < CUTOFF) ? 0.5f * (__cosf(3.14159265358979f * d / CUTOFF) + 1.0f) : 0.0f;
    }

    // -------- GEMM1: [16 x 386] @ We1[386 x 64], K in 3 staged segments + rank-2 tail
    v8f acc[4] = {};
    for (int seg = 0; seg < 3; ++seg) {
        for (int m = 0; m < 16; ++m) {
            const float* rp;
            if (seg == 0)      rp = s + (long long)ssrc[w][m]  * SDIM;
            else if (seg == 1) rp = s + (long long)sdstn[w][m] * SDIM;
            else               rp = e + (long long)(tile*16 + m) * EDIM;
            *(float4*)&buf[w][m][lane * 4] = *(const float4*)(rp + lane * 4);
        }
        const float* B = We1 + seg * SDIM * HID;
        for (int k0 = 0; k0 < SDIM; k0 += 4) {
            int ka = k0 + 2 * half;
            v2f a; a.x = buf[w][n][ka]; a.y = buf[w][n][ka + 1];
            #pragma unroll
            for (int c = 0; c < 4; ++c) {
                v2f b; b.x = B[ka*HID + c*16 + n]; b.y = B[(ka+1)*HID + c*16 + n];
                acc[c] = wmma4(a, b, acc[c]);
            }
        }
    }
    #pragma unroll
    for (int c = 0; c < 4; ++c) {
        int col = c * 16 + n;
        float w384 = We1[384*HID + col];
        float w385 = We1[385*HID + col];
        float bb   = be1[col];
        #pragma unroll
        for (int r = 0; r < 8; ++r) {
            int M = r + 8 * half;
            float x = acc[c][r] + sdist[w][M] * w384 + sead[w][M] * w385 + bb;
            mM[w][M][col] = silu_f(x);
        }
    }

    // -------- GEMM2: [16 x 64] @ We2[64 x 64]
    v8f a2[4] = {};
    for (int k0 = 0; k0 < HID; k0 += 4) {
        int ka = k0 + 2 * half;
        v2f a; a.x = mM[w][n][ka]; a.y = mM[w][n][ka + 1];
        #pragma unroll
        for (int c = 0; c < 4; ++c) {
            v2f b; b.x = We2[ka*HID + c*16 + n]; b.y = We2[(ka+1)*HID + c*16 + n];
            a2[c] = wmma4(a, b, a2[c]);
        }
    }
    #pragma unroll
    for (int c = 0; c < 4; ++c) {
        int col = c * 16 + n;
        float bb = be2[col];
        #pragma unroll
        for (int r = 0; r < 8; ++r) {
            int M = r + 8 * half;
            mM[w][M][col] = silu_f(a2[c][r] + bb) * scw[w][M];   // m tile
        }
    }

    // -------- wv = m @ Wv + bv ; scatter rel*wv into vacc[dst]
    if (lane < 16) {
        float accv = bv[0];
        for (int k = 0; k < HID; ++k) accv += mM[w][lane][k] * Wv[k];
        int d = sdstn[w][lane];
        atomicAdd(&vacc[d*3+0], srel[w][lane][0] * accv);
        atomicAdd(&vacc[d*3+1], srel[w][lane][1] * accv);
        atomicAdd(&vacc[d*3+2], srel[w][lane][2] * accv);
    }

    // -------- GEMM3: e += m @ Weo[64 x 128] + beo  (e_old still staged in buf)
    for (int c = 0; c < 8; ++c) {
        v8f c3 = {};
        for (int k0 = 0; k0 < HID; k0 += 4) {
            int ka = k0 + 2 * half;
            v2f a; a.x = mM[w][n][ka]; a.y = mM[w][n][ka + 1];
            v2f b; b.x = Weo[ka*EDIM + c*16 + n]; b.y = Weo[(ka+1)*EDIM + c*16 + n];
            c3 = wmma4(a, b, c3);
        }
        int col = c * 16 + n;
        float bb = beo[col];
        #pragma unroll
        for (int r = 0; r < 8; ++r) {
            int M = r + 8 * half;
            buf[w][M][col] = buf[w][M][col] + c3[r] + bb;
        }
    }
    for (int m = 0; m < 16; ++m)
        *(float4*)(e + (long long)(tile*16 + m)*EDIM + lane*4) = *(const float4*)&buf[w][m][lane*4];

    // -------- scatter m into agg[dst]
    for (int t = lane; t < 16 * HID; t += 32) {
        int m = t >> 6, k = t & 63;
        atomicAdd(&agg[(long long)sdstn[w][m] * HID + k], mM[w][m][k]);
    }
}

// ---------------------------------------------------------------------------
// Node kernel: one 16-node tile per wave.
//   v += vacc/deg ;  s += silu([s, agg/deg]@Ws1+bs1)@Ws2+bs2
// ---------------------------------------------------------------------------
__global__ __launch_bounds__(WAVES * 32)
void node_kernel(float* __restrict__ s, float* __restrict__ v,
                 const float* __restrict__ Ws1, const float* __restrict__ bs1,
                 const float* __restrict__ Ws2, const float* __restrict__ bs2,
                 const float* __restrict__ agg, const float* __restrict__ vacc,
                 const float* __restrict__ deg, int ntiles)
{
    __shared__ __align__(16) float sbuf[WAVES][16][BSTRIDE];
    __shared__ __align__(16) float abuf[WAVES][16][MSTRIDE];
    __shared__ float sdeg[WAVES][16];

    const int w    = threadIdx.x >> 5;
    const int lane = threadIdx.x & 31;
    const int tile = blockIdx.x * WAVES + w;
    if (tile >= ntiles) return;          // wave-uniform
    const int half = lane >> 4;
    const int n    = lane & 15;

    if (lane < 16) {
        int node = tile * 16 + lane;
        float dm = deg[node]; dm = (dm < 1.0f) ? 1.0f : dm;
        sdeg[w][lane] = dm;
        v[node*3+0] += vacc[node*3+0] / dm;
        v[node*3+1] += vacc[node*3+1] / dm;
        v[node*3+2] += vacc[node*3+2] / dm;
    }

    for (int m = 0; m < 16; ++m)
        *(float4*)&sbuf[w][m][lane*4] = *(const float4*)(s + (long long)(tile*16 + m)*SDIM + lane*4);
    for (int m = 0; m < 16; ++m) {
        if (lane < 16) {
            float4 a4 = *(const float4*)(agg + (long long)(tile*16 + m)*HID + lane*4);
            float dm = sdeg[w][m];
            a4.x /= dm; a4.y /= dm; a4.z /= dm; a4.w /= dm;
            *(float4*)&abuf[w][m][lane*4] = a4;
        }
    }

    // -------- GEMM1: [16 x 192] @ Ws1[192 x 64] (two K-segments)
    v8f acc[4] = {};
    for (int k0 = 0; k0 < SDIM; k0 += 4) {
        int ka = k0 + 2 * half;
        v2f a; a.x = sbuf[w][n][ka]; a.y = sbuf[w][n][ka + 1];
        #pragma unroll
        for (int c = 0; c < 4; ++c) {
            v2f b; b.x = Ws1[ka*HID + c*16 + n]; b.y = Ws1[(ka+1)*HID + c*16 + n];
            acc[c] = wmma4(a, b, acc[c]);
        }
    }
    const float* B2 = Ws1 + SDIM * HID;
    for (int k0 = 0; k0 < HID; k0 += 4) {
        int ka = k0 + 2 * half;
        v2f a; a.x = abuf[w][n][ka]; a.y = abuf[w][n][ka + 1];
        #pragma unroll
        for (int c = 0; c < 4; ++c) {
            v2f b; b.x = B2[ka*HID + c*16 + n]; b.y = B2[(ka+1)*HID + c*16 + n];
            acc[c] = wmma4(a, b, acc[c]);
        }
    }
    #pragma unroll
    for (int c = 0; c < 4; ++c) {
        int col = c * 16 + n;
        float bb = bs1[col];
        #pragma unroll
        for (int r = 0; r < 8; ++r) {
            int M = r + 8 * half;
            abuf[w][M][col] = silu_f(acc[c][r] + bb);   // reuse abuf as transpose buffer
        }
    }

    // -------- GEMM2: s += u @ Ws2[64 x 128] + bs2
    for (int c = 0; c < 8; ++c) {
        v8f c2 = {};
        for (int k0 = 0; k0 < HID; k0 += 4) {
            int ka = k0 + 2 * half;
            v2f a; a.x = abuf[w][n][ka]; a.y = abuf[w][n][ka + 1];
            v2f b; b.x = Ws2[ka*SDIM + c*16 + n]; b.y = Ws2[(ka+1)*SDIM + c*16 + n];
            c2 = wmma4(a, b, c2);
        }
        int col = c * 16 + n;
        float bb = bs2[col];
        #pragma unroll
        for (int r = 0; r < 8; ++r) {
            int M = r + 8 * half;
            sbuf[w][M][col] = sbuf[w][M][col] + c2[r] + bb;
        }
    }
    for (int m = 0; m < 16; ++m)
        *(float4*)(s + (long long)(tile*16 + m)*SDIM + lane*4) = *(const float4*)&sbuf[w][m][lane*4];
}

extern "C" void kernel_launch(void* const* d_in, const int* in_sizes, int n_in,
                              void* d_out, int out_size, void* d_ws, size_t ws_size,
                              hipStream_t stream) {
    (void)n_in; (void)out_size; (void)ws_size;
    const float* s_in = (const float*)d_in[0];
    const float* v_in = (const float*)d_in[1];
    const float* e_in = (const float*)d_in[2];
    const int*   ei   = (const int*)  d_in[3];
    const float* ea_d = (const float*)d_in[4];
    const float* We1  = (const float*)d_in[7];
    const float* be1  = (const float*)d_in[8];
    const float* We2  = (const float*)d_in[9];
    const float* be2  = (const float*)d_in[10];
    const float* Weo  = (const float*)d_in[11];
    const float* beo  = (const float*)d_in[12];
    const float* Wv   = (const float*)d_in[13];
    const float* bv   = (const float*)d_in[14];
    const float* Ws1  = (const float*)d_in[15];
    const float* bs1  = (const float*)d_in[16];
    const float* Ws2  = (const float*)d_in[17];
    const float* bs2  = (const float*)d_in[18];

    const int N = in_sizes[0] / SDIM;
    const int E = in_sizes[2] / EDIM;
    const int* src = ei;
    const int* dst = ei + E;

    float* s = (float*)d_out;
    float* v = s + (size_t)N * SDIM;
    float* e = v + (size_t)N * 3;

    float* deg  = (float*)d_ws;
    float* agg  = deg + N;
    float* vacc = agg + (size_t)N * HID;

    hipMemcpyAsync(s, s_in, (size_t)N * SDIM * sizeof(float), hipMemcpyDeviceToDevice, stream);
    hipMemcpyAsync(v, v_in, (size_t)N * 3    * sizeof(float), hipMemcpyDeviceToDevice, stream);
    hipMemcpyAsync(e, e_in, (size_t)E * EDIM * sizeof(float), hipMemcpyDeviceToDevice, stream);

    {
        long long nz = N;
        zero_kernel<<<(unsigned)((nz + 255) / 256), 256, 0, stream>>>(deg, nz);
    }
    deg_kernel<<<(E + 255) / 256, 256, 0, stream>>>(dst, deg, E);

    const int etiles = E / 16;
    const int ntiles = N / 16;
    for (int i = 0; i < DEPTH; ++i) {
        long long nz = (long long)N * (HID + 3);   // agg and vacc are contiguous
        zero_kernel<<<(unsigned)((nz + 255) / 256), 256, 0, stream>>>(agg, nz);

        edge_kernel<<<(etiles + WAVES - 1) / WAVES, WAVES * 32, 0, stream>>>(
            s, v, e, src, dst, ea_d,
            We1 + (size_t)i * MSG_IN * HID, be1 + i * HID,
            We2 + (size_t)i * HID * HID,    be2 + i * HID,
            Weo + (size_t)i * HID * EDIM,   beo + i * EDIM,
            Wv  + (size_t)i * HID,          bv  + i,
            vacc, agg, etiles);

        node_kernel<<<(ntiles + WAVES - 1) / WAVES, WAVES * 32, 0, stream>>>(
            s, v,
            Ws1 + (size_t)i * SCAL_IN * HID, bs1 + i * HID,
            Ws2 + (size_t)i * HID * SDIM,    bs2 + i * SDIM,
            agg, vacc, deg, ntiles);
    }
}